// MultiViewGCN_75050258530751
// MI455X (gfx1250) — compile-verified
//
#include <hip/hip_runtime.h>

typedef __attribute__((ext_vector_type(2))) float v2f;
typedef __attribute__((ext_vector_type(8))) float v8f;

// ---------------------------------------------------------------------------
// Degree / normalization
// ---------------------------------------------------------------------------
__global__ void k_deg_init(float* __restrict__ deg, int n) {
  int i = blockIdx.x * blockDim.x + threadIdx.x;
  if (i < n) deg[i] = 1.0f;  // self-loop contributes 1 to every node's degree
}

__global__ void k_deg_accum(const int* __restrict__ dst, float* __restrict__ deg, int E) {
  int e = blockIdx.x * blockDim.x + threadIdx.x;
  if (e < E) atomicAdd(&deg[dst[e]], 1.0f);
}

__global__ void k_deg_to_dinv(float* __restrict__ deg, int n) {
  int i = blockIdx.x * blockDim.x + threadIdx.x;
  if (i < n) deg[i] = rsqrtf(fmaxf(deg[i], 1e-12f));
}

// ---------------------------------------------------------------------------
// WMMA GEMM:  H[r][c] = dinv[r] * sum_k X[r][k] * W[k][c]
// Also writes Acc = H (self-loop initialization of the aggregation buffer).
// One wave32 computes one 16-row tile across all FOUT/16 column tiles using
// V_WMMA_F32_16X16X4_F32 (full fp32, matches the reference numerics).
// ---------------------------------------------------------------------------
template <int FIN, int FOUT>
__global__ void k_gemm_wmma(const float* __restrict__ X, const float* __restrict__ W,
                            const float* __restrict__ dinv, float* __restrict__ H,
                            float* __restrict__ Acc, int n) {
  constexpr int NT = FOUT / 16;  // column tiles (<= 4)
  const int wave = threadIdx.x >> 5;
  const int lane = threadIdx.x & 31;
  const int half = lane >> 4;   // 0: lanes 0-15, 1: lanes 16-31
  const int lm   = lane & 15;
  const int m0 = (blockIdx.x * (blockDim.x >> 5) + wave) * 16;
  if (m0 >= n) return;  // wave-uniform exit; EXEC stays all-ones for WMMA

  v8f acc[NT];
#pragma unroll
  for (int t = 0; t < NT; ++t) acc[t] = (v8f)(0.0f);

  // A fragment: lane holds row M = lm; VGPR pair = (K = 2*half, K = 2*half+1)
  int arow = m0 + lm;
  if (arow >= n) arow = n - 1;  // keep EXEC full; (n is a multiple of 16 anyway)
  const float* __restrict__ xr = X + (size_t)arow * FIN;

#pragma unroll
  for (int k0 = 0; k0 < FIN; k0 += 4) {
    v2f a;
    a.x = xr[k0 + 2 * half + 0];
    a.y = xr[k0 + 2 * half + 1];
#pragma unroll
    for (int t = 0; t < NT; ++t) {
      // B fragment: lane holds col N = t*16 + lm; VGPR pair = K = 2*half, 2*half+1
      v2f b;
      b.x = W[(k0 + 2 * half + 0) * FOUT + t * 16 + lm];
      b.y = W[(k0 + 2 * half + 1) * FOUT + t * 16 + lm];
      acc[t] = __builtin_amdgcn_wmma_f32_16x16x4_f32(
          /*neg_a=*/false, a, /*neg_b=*/false, b,
          /*c_mod=*/(short)0, acc[t], /*reuse_a=*/false, /*reuse_b=*/false);
    }
  }

  // Epilogue: C/D VGPR j -> row m0 + j + 8*half, col = tile*16 + lm.
#pragma unroll
  for (int t = 0; t < NT; ++t) {
#pragma unroll
    for (int j = 0; j < 8; ++j) {
      int r = m0 + j + 8 * half;
      if (r < n) {
        float v = acc[t][j] * dinv[r];
        size_t off = (size_t)r * FOUT + t * 16 + lm;
        H[off]   = v;
        Acc[off] = v;
      }
    }
  }
}

// ---------------------------------------------------------------------------
// Edge scatter: Acc[dst][f] += H[src][f]  (H already scaled by dinv[src]).
// Consecutive threads cover consecutive features of one edge -> coalesced
// gather and same-cacheline atomics.
// ---------------------------------------------------------------------------
__global__ void k_edge_scatter(const int* __restrict__ src, const int* __restrict__ dst,
                               const float* __restrict__ H, float* __restrict__ Acc,
                               int E, int fshift) {
  long long tid = (long long)blockIdx.x * blockDim.x + threadIdx.x;
  const int fout = 1 << fshift;
  if (tid >= ((long long)E << fshift)) return;
  int f = (int)(tid & (fout - 1));
  int e = (int)(tid >> fshift);
  int s = src[e];
  int d = dst[e];
  atomicAdd(&Acc[d * fout + f], H[s * fout + f]);
}

// ---------------------------------------------------------------------------
// Finalize (in place): Acc[i][f] = relu(dinv[i] * Acc[i][f] + b[f])
// ---------------------------------------------------------------------------
__global__ void k_finalize(float* __restrict__ Acc, const float* __restrict__ dinv,
                           const float* __restrict__ bias, int n, int fout) {
  long long tid = (long long)blockIdx.x * blockDim.x + threadIdx.x;
  if (tid >= (long long)n * fout) return;
  int f = (int)(tid % fout);
  int i = (int)(tid / fout);
  float v = dinv[i] * Acc[tid] + bias[f];
  Acc[tid] = fmaxf(v, 0.0f);
}

// ---------------------------------------------------------------------------
// Pooling
// ---------------------------------------------------------------------------
__global__ void k_zero(float* __restrict__ p, int n) {
  int i = blockIdx.x * blockDim.x + threadIdx.x;
  if (i < n) p[i] = 0.0f;
}

__global__ void k_pool_accum(const float* __restrict__ X, const int* __restrict__ batch,
                             float* __restrict__ out, int n, int fout) {
  long long tid = (long long)blockIdx.x * blockDim.x + threadIdx.x;
  if (tid >= (long long)n * fout) return;
  int f = (int)(tid % fout);
  int i = (int)(tid / fout);
  atomicAdd(&out[batch[i] * fout + f], X[tid]);
}

__global__ void k_pool_count(const int* __restrict__ batch, float* __restrict__ cnt, int n) {
  int i = blockIdx.x * blockDim.x + threadIdx.x;
  if (i < n) atomicAdd(&cnt[batch[i]], 1.0f);
}

__global__ void k_pool_div(float* __restrict__ out, const float* __restrict__ cnt,
                           int G, int fout) {
  int tid = blockIdx.x * blockDim.x + threadIdx.x;
  if (tid >= G * fout) return;
  int g = tid / fout;
  out[tid] = out[tid] / fmaxf(cnt[g], 1.0f);
}

// ---------------------------------------------------------------------------
// Host driver
// ---------------------------------------------------------------------------
static inline unsigned blocks_for(long long work, int bs) {
  return (unsigned)((work + bs - 1) / bs);
}

extern "C" void kernel_launch(void* const* d_in, const int* in_sizes, int n_in,
                              void* d_out, int out_size, void* d_ws, size_t ws_size,
                              hipStream_t stream) {
  (void)n_in; (void)ws_size;
  constexpr int F_IN = 128, H1 = 32, H2 = 64;

  const float* xs[3]  = {(const float*)d_in[0], (const float*)d_in[3], (const float*)d_in[6]};
  const int*   eis[3] = {(const int*)d_in[1],   (const int*)d_in[4],   (const int*)d_in[7]};
  const int*   bts[3] = {(const int*)d_in[2],   (const int*)d_in[5],   (const int*)d_in[8]};
  const float* W1 = (const float*)d_in[9];
  const float* b1 = (const float*)d_in[10];
  const float* W2 = (const float*)d_in[11];
  const float* b2 = (const float*)d_in[12];
  const float* W3 = (const float*)d_in[13];
  const float* b3 = (const float*)d_in[14];

  const int n = in_sizes[0] / F_IN;      // 100000
  const int E = in_sizes[1] / 2;         // 1600000
  const int G = out_size / (3 * H2);     // 512

  // Workspace carve-up
  char* ws = (char*)d_ws;
  float* dinv = (float*)ws; ws += (size_t)n * sizeof(float);
  float* bufP = (float*)ws; ws += (size_t)n * H2 * sizeof(float);
  float* bufQ = (float*)ws; ws += (size_t)n * H2 * sizeof(float);
  float* bufR = (float*)ws; ws += (size_t)n * H2 * sizeof(float);
  float* cnt  = (float*)ws; ws += (size_t)G * sizeof(float);

  const int BS = 256;
  const int tiles = (n + 15) / 16;
  const unsigned gemm_blocks = (unsigned)((tiles + 7) / 8);  // 8 waves per block

  for (int v = 0; v < 3; ++v) {
    const float* x   = xs[v];
    const int*   src = eis[v];
    const int*   dst = eis[v] + E;
    const int*   bat = bts[v];
    float*       out = (float*)d_out + (size_t)v * G * H2;

    // deg -> dinv
    k_deg_init<<<blocks_for(n, BS), BS, 0, stream>>>(dinv, n);
    k_deg_accum<<<blocks_for(E, BS), BS, 0, stream>>>(dst, dinv, E);
    k_deg_to_dinv<<<blocks_for(n, BS), BS, 0, stream>>>(dinv, n);

    // Layer 1: x[n,128] @ W1[128,32] -> bufQ (h'), bufR (acc)
    k_gemm_wmma<F_IN, H1><<<gemm_blocks, BS, 0, stream>>>(x, W1, dinv, bufQ, bufR, n);
    k_edge_scatter<<<blocks_for((long long)E * H1, BS), BS, 0, stream>>>(src, dst, bufQ, bufR, E, 5);
    k_finalize<<<blocks_for((long long)n * H1, BS), BS, 0, stream>>>(bufR, dinv, b1, n, H1);

    // Layer 2: bufR[n,32] @ W2[32,64] -> bufQ (h'), bufP (acc)
    k_gemm_wmma<H1, H2><<<gemm_blocks, BS, 0, stream>>>(bufR, W2, dinv, bufQ, bufP, n);
    k_edge_scatter<<<blocks_for((long long)E * H2, BS), BS, 0, stream>>>(src, dst, bufQ, bufP, E, 6);
    k_finalize<<<blocks_for((long long)n * H2, BS), BS, 0, stream>>>(bufP, dinv, b2, n, H2);

    // Layer 3: bufP[n,64] @ W3[64,64] -> bufQ (h'), bufR (acc)
    k_gemm_wmma<H2, H2><<<gemm_blocks, BS, 0, stream>>>(bufP, W3, dinv, bufQ, bufR, n);
    k_edge_scatter<<<blocks_for((long long)E * H2, BS), BS, 0, stream>>>(src, dst, bufQ, bufR, E, 6);
    k_finalize<<<blocks_for((long long)n * H2, BS), BS, 0, stream>>>(bufR, dinv, b3, n, H2);

    // Global mean pool over graphs
    k_zero<<<blocks_for(G * H2, BS), BS, 0, stream>>>(out, G * H2);
    k_zero<<<blocks_for(G, BS), BS, 0, stream>>>(cnt, G);
    k_pool_accum<<<blocks_for((long long)n * H2, BS), BS, 0, stream>>>(bufR, bat, out, n, H2);
    k_pool_count<<<blocks_for(n, BS), BS, 0, stream>>>(bat, cnt, n);
    k_pool_div<<<blocks_for(G * H2, BS), BS, 0, stream>>>(out, cnt, G, H2);
  }
}